// RankListNetNode_34600256537541
// MI455X (gfx1250) — compile-verified
//
#include <hip/hip_runtime.h>

// ---------------------------------------------------------------------------
// Hetero-GraphSAGE (2 layers) for MI455X / gfx1250.
// Scatter-mean via global f32 atomics (L2-resident), dense GEMMs via
// V_WMMA_F32_16X16X4_F32 (fp32-exact matrix pipe). Output reshape is identity.
// ---------------------------------------------------------------------------

typedef float v2f __attribute__((ext_vector_type(2)));
typedef float v8f __attribute__((ext_vector_type(8)));

#define NVN 100000   // var nodes
#define NCN 50000    // config nodes
#define NED 800000   // edges per direction
#define DVF 64       // var feature dim
#define DCF 16       // config feature dim
#define DHF 128      // hidden dim
#define DOF 64       // out dim

// ---------------- zero fill ----------------
__global__ void __launch_bounds__(256) zero_f32(float* __restrict__ p, int n) {
  int i = blockIdx.x * 256 + threadIdx.x;
  if (i < n) p[i] = 0.0f;
}

// ---------------- scatter-mean accumulation ----------------
// one thread per (edge, 4-float chunk); F = F4*4 features per node
template <int F4>
__global__ void __launch_bounds__(256) scatter_mean_k(
    const float* __restrict__ xsrc, const int* __restrict__ src,
    const int* __restrict__ dst, float* __restrict__ msum,
    float* __restrict__ cnt, int nEdges) {
  int t = blockIdx.x * 256 + threadIdx.x;
  int total = nEdges * F4;
  if (t >= total) return;
  int e = t / F4;
  int c = t - e * F4;
  int s = src[e];
  int d = dst[e];
  const int F = F4 * 4;
  float4 v = *reinterpret_cast<const float4*>(xsrc + (size_t)s * F + c * 4);
  float* o = msum + (size_t)d * F + c * 4;
  unsafeAtomicAdd(o + 0, v.x);
  unsafeAtomicAdd(o + 1, v.y);
  unsafeAtomicAdd(o + 2, v.z);
  unsafeAtomicAdd(o + 3, v.w);
  if (cnt != nullptr && c == 0) unsafeAtomicAdd(cnt + d, 1.0f);
}

// ---------------- fp32 WMMA helper ----------------
__device__ __forceinline__ v8f wmma4(v2f a, v2f b, v8f c) {
  // D = A(16x4,f32) * B(4x16,f32) + C(16x16,f32)
  return __builtin_amdgcn_wmma_f32_16x16x4_f32(
      /*neg_a=*/false, a, /*neg_b=*/false, b,
      /*c_mod=*/(short)0, c, /*reuse_a=*/false, /*reuse_b=*/false);
}

// ---------------- fused SAGE GEMM ----------------
// out[M,N] = act( (msum/max(cnt,1))[M,K1] @ W1[K1,N] + xdst[M,K2] @ W2[K2,N] + bias[N] )
// N = NTILES*16. 256 threads = 8 waves; each wave owns a 16-row strip.
template <int NTILES, bool RELU>
__global__ void __launch_bounds__(256) sage_gemm_k(
    const float* __restrict__ msum, const float* __restrict__ cnt,
    const float* __restrict__ W1, int K1,
    const float* __restrict__ xdst, const float* __restrict__ W2, int K2,
    const float* __restrict__ bias, float* __restrict__ out, int M) {
  const int N = NTILES * 16;
  int lane  = threadIdx.x & 31;
  int wave  = threadIdx.x >> 5;
  int mBase = (blockIdx.x * 8 + wave) * 16;
  if (mBase >= M) return;          // wave-uniform: EXEC stays all-ones for WMMA

  int mrow = lane & 15;            // M (for A) / N (for B) index within tile
  int half = lane >> 4;            // lane half selects K pair {0,1} vs {2,3}
  int kh   = half * 2;
  int row  = mBase + mrow;

  float inv = 1.0f / fmaxf(cnt[row], 1.0f);

  v8f acc[NTILES] = {};

  // term 1: mean @ W1  (mean applied on the fly)
  const float* arow = msum + (size_t)row * K1;
  for (int k = 0; k < K1; k += 4) {
    v2f a;
    a.x = arow[k + kh] * inv;
    a.y = arow[k + kh + 1] * inv;
#pragma unroll
    for (int j = 0; j < NTILES; ++j) {
      v2f b;
      b.x = W1[(size_t)(k + kh) * N + j * 16 + mrow];
      b.y = W1[(size_t)(k + kh + 1) * N + j * 16 + mrow];
      acc[j] = wmma4(a, b, acc[j]);
    }
  }
  // term 2: x_dst @ W2
  const float* xrow = xdst + (size_t)row * K2;
  for (int k = 0; k < K2; k += 4) {
    v2f a;
    a.x = xrow[k + kh];
    a.y = xrow[k + kh + 1];
#pragma unroll
    for (int j = 0; j < NTILES; ++j) {
      v2f b;
      b.x = W2[(size_t)(k + kh) * N + j * 16 + mrow];
      b.y = W2[(size_t)(k + kh + 1) * N + j * 16 + mrow];
      acc[j] = wmma4(a, b, acc[j]);
    }
  }

  // epilogue: bias (+relu) and store.  D layout: VGPR v holds M = v + 8*half.
#pragma unroll
  for (int j = 0; j < NTILES; ++j) {
    int col  = j * 16 + mrow;
    float bv = bias[col];
#pragma unroll
    for (int v = 0; v < 8; ++v) {
      int m = mBase + (half << 3) + v;
      float val = acc[j][v] + bv;
      if (RELU) val = fmaxf(val, 0.0f);
      out[(size_t)m * N + col] = val;
    }
  }
}

// ---------------------------------------------------------------------------
extern "C" void kernel_launch(void* const* d_in, const int* in_sizes, int n_in,
                              void* d_out, int out_size, void* d_ws, size_t ws_size,
                              hipStream_t stream) {
  const float* x_var    = (const float*)d_in[0];
  const float* x_config = (const float*)d_in[1];
  const int*   src_v2c  = (const int*)d_in[2];
  const int*   dst_v2c  = (const int*)d_in[3];
  const int*   src_c2v  = (const int*)d_in[4];
  const int*   dst_c2v  = (const int*)d_in[5];
  // d_in[6] batch_config: unused — uniform contiguous batches make the final
  // to_dense_batch an identity reshape of o_config.
  const float* Wl1_v2c = (const float*)d_in[7];
  const float* bl1_v2c = (const float*)d_in[8];
  const float* Wr1_v2c = (const float*)d_in[9];
  const float* Wl1_c2v = (const float*)d_in[10];
  const float* bl1_c2v = (const float*)d_in[11];
  const float* Wr1_c2v = (const float*)d_in[12];
  const float* Wl2     = (const float*)d_in[13];
  const float* bl2     = (const float*)d_in[14];
  const float* Wr2     = (const float*)d_in[15];

  float* ws        = (float*)d_ws;
  float* msum_v2c  = ws;                                  // NC*64
  float* cnt_v2c   = msum_v2c + (size_t)NCN * DVF;        // NC
  float* msum_c2v  = cnt_v2c + NCN;                       // NV*16
  float* cnt_c2v   = msum_c2v + (size_t)NVN * DCF;        // NV
  float* msum2     = cnt_c2v + NVN;                       // NC*128
  float* h_config  = msum2 + (size_t)NCN * DHF;           // NC*128
  float* h_var     = h_config + (size_t)NCN * DHF;        // NV*128

  // zero all atomic accumulators (they are laid out contiguously first)
  int nZero = NCN * DVF + NCN + NVN * DCF + NVN + NCN * DHF;
  zero_f32<<<(nZero + 255) / 256, 256, 0, stream>>>(ws, nZero);

  // layer-1 scatter means
  scatter_mean_k<DVF / 4><<<(NED * (DVF / 4) + 255) / 256, 256, 0, stream>>>(
      x_var, src_v2c, dst_v2c, msum_v2c, cnt_v2c, NED);
  scatter_mean_k<DCF / 4><<<(NED * (DCF / 4) + 255) / 256, 256, 0, stream>>>(
      x_config, src_c2v, dst_c2v, msum_c2v, cnt_c2v, NED);

  // layer-1 fused GEMMs (relu)
  sage_gemm_k<DHF / 16, true><<<(NCN / 16 + 7) / 8, 256, 0, stream>>>(
      msum_v2c, cnt_v2c, Wl1_v2c, DVF, x_config, Wr1_v2c, DCF, bl1_v2c,
      h_config, NCN);
  sage_gemm_k<DHF / 16, true><<<(NVN / 16 + 7) / 8, 256, 0, stream>>>(
      msum_c2v, cnt_c2v, Wl1_c2v, DCF, x_var, Wr1_c2v, DVF, bl1_c2v,
      h_var, NVN);

  // layer-2 scatter (same dst indices as layer-1 v2c -> reuse cnt_v2c)
  scatter_mean_k<DHF / 4><<<(NED * (DHF / 4) + 255) / 256, 256, 0, stream>>>(
      h_var, src_v2c, dst_v2c, msum2, nullptr, NED);

  // layer-2 GEMM straight into d_out (to_dense_batch == reshape here)
  sage_gemm_k<DOF / 16, false><<<(NCN / 16 + 7) / 8, 256, 0, stream>>>(
      msum2, cnt_v2c, Wl2, DHF, h_config, Wr2, DHF, bl2, (float*)d_out, NCN);
}